// QuantumInspiredAttention_7344394076254
// MI455X (gfx1250) — compile-verified
//
#include <hip/hip_runtime.h>
#include <hip/hip_bf16.h>

typedef float v2f __attribute__((ext_vector_type(2)));
typedef float v8f __attribute__((ext_vector_type(8)));

#define BQ 16
#define SQ 512
#define DQ 1024
#define NCH 8            // S split into 8 chunks for the mean reduction
#define SCH (SQ / NCH)   // 64 rows per chunk

// ---------------------------------------------------------------------------
// Stage 1a: partial column sums of x over the sequence axis.
// grid = (B, NCH), block = 256.  Thread t owns float4 of columns d = 4t..4t+3.
// Fully coalesced float4 loads (D*4 bytes row stride, 16B aligned).
// ---------------------------------------------------------------------------
__global__ __launch_bounds__(256) void qia_mean_partial(
    const float* __restrict__ x, float* __restrict__ partial)
{
    const int b = blockIdx.x, ch = blockIdx.y, tid = threadIdx.x;
    const float4* xp = (const float4*)x + (size_t)(b * SQ + ch * SCH) * (DQ / 4) + tid;
    float4 acc = {0.f, 0.f, 0.f, 0.f};
#pragma unroll 4
    for (int s = 0; s < SCH; ++s) {
        float4 v = xp[(size_t)s * (DQ / 4)];
        acc.x += v.x; acc.y += v.y; acc.z += v.z; acc.w += v.w;
    }
    ((float4*)partial)[(b * NCH + ch) * (DQ / 4) + tid] = acc;
}

// Stage 1b: reduce the NCH partials and scale by 1/S  ->  xbar[b,d]
__global__ __launch_bounds__(256) void qia_mean_finish(
    const float* __restrict__ partial, float* __restrict__ xbar)
{
    const int b = blockIdx.x, tid = threadIdx.x;
    float4 acc = {0.f, 0.f, 0.f, 0.f};
#pragma unroll
    for (int ch = 0; ch < NCH; ++ch) {
        float4 v = ((const float4*)partial)[(b * NCH + ch) * (DQ / 4) + tid];
        acc.x += v.x; acc.y += v.y; acc.z += v.z; acc.w += v.w;
    }
    const float inv = 1.0f / (float)SQ;
    acc.x *= inv; acc.y *= inv; acc.z *= inv; acc.w *= inv;
    ((float4*)xbar)[b * (DQ / 4) + tid] = acc;
}

// ---------------------------------------------------------------------------
// Stage 2/3: Y[16,1024] = A[16,1024] @ W[1024,1024] + bias, full fp32 via
// V_WMMA_F32_16X16X4_F32.  One wave per 16x16 output tile, grid = 64 waves.
//
// A-fragment (16x4 f32, 2 VGPRs): lanes 0-15 -> row M=lane, K = k+{0,1};
//                                 lanes 16-31 -> row M=lane-16, K = k+{2,3}.
// B-fragment (4x16 f32, 2 VGPRs): mirrored -> lane half selects K pair,
//                                 vector element selects K within the pair,
//                                 N = lane & 15.
// C/D (16x16 f32, 8 VGPRs): element r of lane L is (M = r + 8*(L>=16), N = L&15).
// EXEC is all-ones: 32-thread block, straight-line code, no divergence.
// ---------------------------------------------------------------------------
__global__ __launch_bounds__(32) void qia_gemm16_wmma(
    const float* __restrict__ A,    // [16 x DQ] row-major
    const float* __restrict__ W,    // [DQ x DQ] row-major
    const float* __restrict__ bias, // [DQ]
    float* __restrict__ Y)          // [16 x DQ] row-major
{
    const int lane = threadIdx.x;
    const int half = lane >> 4;   // 0 | 1  (selects K pair)
    const int lm   = lane & 15;
    const int n0   = blockIdx.x * 16;
    const int col  = n0 + lm;

    const float* arow = A + (size_t)lm * DQ + 2 * half; // A[lm][k + 2*half + {0,1}]
    const float* wcol = W + (size_t)(2 * half) * DQ + col; // W[k + 2*half + {0,1}][col]

    v8f c = {};
#pragma unroll 8
    for (int k = 0; k < DQ; k += 4) {
        v2f a, b;
        a.x = arow[k];
        a.y = arow[k + 1];
        b.x = wcol[(size_t)k * DQ];
        b.y = wcol[(size_t)(k + 1) * DQ];
        // 8 args: (neg_a, A, neg_b, B, c_mod, C, reuse_a, reuse_b)
        c = __builtin_amdgcn_wmma_f32_16x16x4_f32(
            false, a, false, b, (short)0, c, false, false);
    }

    const float bb = bias[col];
#pragma unroll
    for (int r = 0; r < 8; ++r) {
        const int row = r + 8 * half;
        Y[(size_t)row * DQ + col] = c[r] + bb;
    }
}

// ---------------------------------------------------------------------------
// Stage 4: broadcast y2[b,:] to out[b,s,:].  32 MB of float4 stores; y2 is
// 64 KB and stays resident in L2 for the reads.
// ---------------------------------------------------------------------------
__global__ __launch_bounds__(256) void qia_broadcast(
    const float4* __restrict__ y2, float4* __restrict__ out)
{
    const unsigned i = blockIdx.x * 256u + threadIdx.x; // < B*S*D/4 = 2^21
    const unsigned b  = i >> 17;   // S*D/4 = 131072 = 2^17
    const unsigned d4 = i & 255u;  // D/4 = 256
    out[i] = y2[(b << 8) + d4];
}

// ---------------------------------------------------------------------------
// Input dict order: x, wq, bq, wk, bk, wv, bv, wo, bo, w1, b1, w2, b2,
//                   hadamard, rot.  Only x, wv, bv, wo, bo matter (the
//                   softmax(s/s) term makes attention uniform = 1/S).
// ---------------------------------------------------------------------------
extern "C" void kernel_launch(void* const* d_in, const int* in_sizes, int n_in,
                              void* d_out, int out_size, void* d_ws, size_t ws_size,
                              hipStream_t stream)
{
    const float* x  = (const float*)d_in[0];
    const float* wv = (const float*)d_in[5];
    const float* bv = (const float*)d_in[6];
    const float* wo = (const float*)d_in[7];
    const float* bo = (const float*)d_in[8];
    float* out = (float*)d_out;

    char* ws = (char*)d_ws;
    float* partial = (float*)ws;                                  // 16*8*1024 f32 = 512 KB
    float* xbar    = (float*)(ws + (size_t)BQ * NCH * DQ * 4);    // 64 KB
    float* y1      = xbar + BQ * DQ;                              // 64 KB
    float* y2      = y1 + BQ * DQ;                                // 64 KB

    qia_mean_partial<<<dim3(BQ, NCH), 256, 0, stream>>>(x, partial);
    qia_mean_finish<<<BQ, 256, 0, stream>>>(partial, xbar);
    qia_gemm16_wmma<<<DQ / 16, 32, 0, stream>>>(xbar, wv, bv, y1);
    qia_gemm16_wmma<<<DQ / 16, 32, 0, stream>>>(y1, wo, bo, y2);
    qia_broadcast<<<(BQ * SQ * DQ / 4) / 256, 256, 0, stream>>>(
        (const float4*)y2, (float4*)out);
}